// TopKGate_79319456022679
// MI455X (gfx1250) — compile-verified
//
#include <hip/hip_runtime.h>
#include <stdint.h>

// Problem constants (static in the reference)
#define B_ROWS  8192
#define N_COLS  16384
#define K_TOP   512
#define THREADS 256
#define NWAVES  (THREADS / 32)              // 8 wave32s
#define EPT     (N_COLS / THREADS)          // 64 elements per thread
#define VEC     4
#define NCHUNK  (N_COLS / (THREADS * VEC))  // 16 float4 chunks per row

typedef uint32_t u32x4 __attribute__((ext_vector_type(4)));
typedef int      i32x4 __attribute__((ext_vector_type(4)));
typedef int      i32x8 __attribute__((ext_vector_type(8)));
typedef float    f32x4 __attribute__((ext_vector_type(4)));   // native vector: OK for NT store

// ReLU'd non-negative floats compare identically to their uint bit patterns.
__device__ __forceinline__ uint32_t relu_bits(float x) {
  return __float_as_uint(fmaxf(x, 0.0f));
}

#if defined(__has_builtin)
#if __has_builtin(__builtin_amdgcn_tensor_load_to_lds)
#define USE_TDM 1
#endif
#if __has_builtin(__builtin_amdgcn_ballot_w32)
#define USE_BALLOT 1
#endif
#endif

#ifdef USE_TDM
// One TDM descriptor: 2D tile (16384 x 1) of 4-byte elements, global -> LDS.
__device__ __forceinline__ void tdm_load_row_to_lds(uint32_t lds_addr, const float* gptr) {
  const uint64_t ga = (uint64_t)(uintptr_t)gptr;
  u32x4 g0;
  g0.x = 1u;                                           // count=1 (user mode, load)
  g0.y = lds_addr;                                     // D# lds_addr      [63:32]
  g0.z = (uint32_t)ga;                                 // global_addr lo   [95:64]
  g0.w = ((uint32_t)(ga >> 32) & 0x01FFFFFFu)          // global_addr[56:32]
         | (2u << 30);                                 // type=2 ("image") [127:126]
  i32x8 g1;
  g1[0] = (int)(2u << 16);                             // data_size=2 -> 4 bytes; no multicast/pad
  g1[1] = (int)((uint32_t)N_COLS << 16);               // tensor_dim0[15:0] @ [63:48]
  g1[2] = (int)(((uint32_t)N_COLS >> 16) | (1u << 16));// tensor_dim0 hi | tensor_dim1=1
  g1[3] = (int)((uint32_t)N_COLS << 16);               // tensor_dim1 hi=0 | tile_dim0=16384
  g1[4] = 1;                                           // tile_dim1=1, tile_dim2=0 (unused)
  g1[5] = N_COLS;                                      // tensor_dim0_stride lo
  g1[6] = 0;                                           // stride hi / dim1_stride lo
  g1[7] = 0;                                           // dim1_stride hi
  const i32x4 gz = {0, 0, 0, 0};                       // groups 2/3: no 3D+, tile_dim3/4=0
#if __clang_major__ >= 23
  const i32x8 gz8 = {0, 0, 0, 0, 0, 0, 0, 0};
  __builtin_amdgcn_tensor_load_to_lds(g0, g1, gz, gz, gz8, 0);
#else
  __builtin_amdgcn_tensor_load_to_lds(g0, g1, gz, gz, 0);
#endif
}
#endif

extern "C" __global__ __launch_bounds__(THREADS)
void topk_sparse_kernel(const float* __restrict__ h,
                        float* __restrict__ out_sparse,
                        int* __restrict__ out_idx) {
  extern __shared__ float s_row[];                 // 64 KB dynamic LDS: the whole row
  __shared__ uint32_t s_hist[NWAVES][256];         // per-wave radix histograms (8 KB)
  __shared__ uint32_t s_scan[THREADS];             // scan / wave-total workspace
  __shared__ uint64_t s_key[K_TOP];                // (bits<<32)|~idx winner keys
  __shared__ uint32_t s_bitmap[N_COLS / 32];       // keep-flags, 2 KB
  __shared__ uint32_t s_sel[4];                    // 0:prefix 1:need 2:append counter

  const int tid  = threadIdx.x;
  const int lane = tid & 31;
  const int wave = tid >> 5;
  const int row  = blockIdx.x;
  const float* rowp = h + (size_t)row * N_COLS;
  // Generic pointer low 32 bits == LDS byte offset (LDS aperture truncation).
  const uint32_t lds0 = (uint32_t)(uintptr_t)(void*)&s_row[0];

  // ---- Stage row into LDS: Tensor Data Mover (TENSORcnt), fallback ASYNC path ----
#ifdef USE_TDM
  if (__builtin_amdgcn_readfirstlane(tid) < 32) {  // uniform: only wave 0 issues/waits
    if (lane == 0) tdm_load_row_to_lds(lds0, rowp);
    __builtin_amdgcn_s_wait_tensorcnt(0);
  }
#else
  {
    const uint64_t gbase = (uint64_t)(uintptr_t)rowp;
#pragma unroll
    for (int j = 0; j < NCHUNK; ++j) {
      const uint32_t off = (uint32_t)(j * THREADS + tid) * 16u;  // bytes
      const uint32_t lds_addr = lds0 + off;
      asm volatile("global_load_async_to_lds_b128 %0, %1, %2"
                   :: "v"(lds_addr), "v"(off), "s"(gbase) : "memory");
    }
    asm volatile("s_wait_asynccnt 0" ::: "memory");
  }
#endif

  // Metadata init (DMA in flight until the barrier).
#pragma unroll
  for (int w = 0; w < NWAVES; ++w) s_hist[w][tid] = 0u;
  s_bitmap[tid] = 0u;
  s_bitmap[tid + THREADS] = 0u;
  if (tid == 0) { s_sel[0] = 0u; s_sel[1] = K_TOP; s_sel[2] = 0u; }
  __syncthreads();

  // ---- 4-pass MSD radix select for the K-th largest value ----
  uint32_t prefix = 0u, need = K_TOP;
#pragma unroll 1
  for (int pass = 0; pass < 4; ++pass) {
    const int shift = 24 - pass * 8;
    const uint32_t mask_hi = (pass == 0) ? 0u : (0xFFFFFFFFu << (shift + 8));
#pragma unroll 4
    for (int j = 0; j < EPT; ++j) {
      const uint32_t b = relu_bits(s_row[j * THREADS + tid]);
      if ((b & mask_hi) == prefix)
        atomicAdd(&s_hist[wave][(b >> shift) & 0xFFu], 1u);  // per-wave: low contention
    }
    __syncthreads();
    // Column-reduce the 8 sub-histograms into row 0 (thread t owns bin t).
    {
      uint32_t sum = 0u;
#pragma unroll
      for (int w = 0; w < NWAVES; ++w) sum += s_hist[w][tid];
      s_hist[0][tid] = sum;
    }
    __syncthreads();
    if (tid == 0) {
      uint32_t cum = 0u;
      int d = 255;
      for (; d > 0; --d) {                 // scan from largest digit down
        const uint32_t c = s_hist[0][d];
        if (cum + c >= need) break;
        cum += c;
      }
      s_sel[0] = prefix | ((uint32_t)d << shift);
      s_sel[1] = need - cum;               // still needed within chosen digit
    }
    __syncthreads();
    prefix = s_sel[0];
    need   = s_sel[1];
#pragma unroll
    for (int w = 0; w < NWAVES; ++w) s_hist[w][tid] = 0u;  // clear for next pass
    __syncthreads();
  }
  const uint32_t T = prefix;               // exact threshold bits
  const uint32_t tiesNeeded = need;        // # of ==T kept, lowest index first

  // ---- Collect winners: >T unordered append; ==T ranked in index order ----
  uint32_t running = 0u;                   // equals seen in earlier chunks (uniform)
#pragma unroll 1
  for (int j = 0; j < NCHUNK; ++j) {
    const int base = j * THREADS * VEC + tid * VEC;
    const f32x4 f4 = *(const f32x4*)&s_row[base];
    uint32_t bts[VEC] = { relu_bits(f4.x), relu_bits(f4.y),
                          relu_bits(f4.z), relu_bits(f4.w) };
    uint32_t eqc = 0u;
#pragma unroll
    for (int v = 0; v < VEC; ++v) eqc += (bts[v] == T) ? 1u : 0u;

    uint32_t excl, total;   // exclusive prefix of eqc over the block, and block total
#ifdef USE_BALLOT
    {
      // wave32 ballot scan: eqc in [0,4] -> 3 bit-planes
      const uint32_t b0 = __builtin_amdgcn_ballot_w32((eqc >> 0) & 1u);
      const uint32_t b1 = __builtin_amdgcn_ballot_w32((eqc >> 1) & 1u);
      const uint32_t b2 = __builtin_amdgcn_ballot_w32((eqc >> 2) & 1u);
      const uint32_t lt = (1u << lane) - 1u;
      const uint32_t in_wave = __popc(b0 & lt) + 2u * __popc(b1 & lt) + 4u * __popc(b2 & lt);
      const uint32_t wtotal  = __popc(b0) + 2u * __popc(b1) + 4u * __popc(b2);
      if (lane == 0) s_scan[wave] = wtotal;
      __syncthreads();
      uint32_t wbase = 0u; total = 0u;
#pragma unroll
      for (int w = 0; w < NWAVES; ++w) {
        const uint32_t c = s_scan[w];
        wbase += (w < wave) ? c : 0u;
        total += c;
      }
      __syncthreads();
      excl = wbase + in_wave;
    }
#else
    {
      s_scan[tid] = eqc;
      __syncthreads();
#pragma unroll
      for (int off = 1; off < THREADS; off <<= 1) {
        uint32_t vv = 0u;
        if (tid >= off) vv = s_scan[tid - off];
        __syncthreads();
        if (tid >= off) s_scan[tid] += vv;
        __syncthreads();
      }
      excl  = s_scan[tid] - eqc;
      total = s_scan[THREADS - 1];
      __syncthreads();
    }
#endif
    uint32_t rank = running + excl;        // rank in global index order among ==T
    running += total;

#pragma unroll
    for (int v = 0; v < VEC; ++v) {
      const int i = base + v;
      bool keep = false;
      if (bts[v] > T) keep = true;
      else if (bts[v] == T) { keep = (rank < tiesNeeded); ++rank; }
      if (keep) {
        const uint32_t pos = atomicAdd(&s_sel[2], 1u);
        s_key[pos] = ((uint64_t)bts[v] << 32) | (uint32_t)~(uint32_t)i;
        atomicOr(&s_bitmap[i >> 5], 1u << (i & 31));
      }
    }
  }
  __syncthreads();

  // ---- Bitonic sort of 512 keys, descending (value desc, index asc on ties) ----
#pragma unroll 1
  for (int kk = 2; kk <= K_TOP; kk <<= 1) {
#pragma unroll 1
    for (int jj = kk >> 1; jj > 0; jj >>= 1) {
#pragma unroll
      for (int e = 0; e < K_TOP / THREADS; ++e) {
        const int i = e * THREADS + tid;
        const int ixj = i ^ jj;
        if (ixj > i) {
          const uint64_t a = s_key[i], b2 = s_key[ixj];
          const bool up = ((i & kk) == 0);
          if (up ? (a < b2) : (a > b2)) { s_key[i] = b2; s_key[ixj] = a; }
        }
      }
      __syncthreads();
    }
  }

  // ---- Emit sorted top-k indices ----
#pragma unroll
  for (int e = 0; e < K_TOP / THREADS; ++e) {
    const int i = e * THREADS + tid;
    const int idx = (int)~((uint32_t)s_key[i]);
    __builtin_nontemporal_store(idx, &out_idx[(size_t)row * K_TOP + i]);
  }

  // ---- Emit dense sparse row: kept values else 0 (streaming, non-temporal) ----
#pragma unroll 1
  for (int j = 0; j < NCHUNK; ++j) {
    const int base = j * THREADS * VEC + tid * VEC;
    const f32x4 f4 = *(const f32x4*)&s_row[base];
    const uint32_t bm = s_bitmap[base >> 5] >> (base & 31);
    f32x4 o;
    o.x = (bm & 1u) ? fmaxf(f4.x, 0.0f) : 0.0f;
    o.y = (bm & 2u) ? fmaxf(f4.y, 0.0f) : 0.0f;
    o.z = (bm & 4u) ? fmaxf(f4.z, 0.0f) : 0.0f;
    o.w = (bm & 8u) ? fmaxf(f4.w, 0.0f) : 0.0f;
    __builtin_nontemporal_store(o, (f32x4*)(out_sparse + (size_t)row * N_COLS + base));
  }
}

extern "C" void kernel_launch(void* const* d_in, const int* in_sizes, int n_in,
                              void* d_out, int out_size, void* d_ws, size_t ws_size,
                              hipStream_t stream) {
  (void)in_sizes; (void)n_in; (void)d_ws; (void)ws_size; (void)out_size;
  const float* h = (const float*)d_in[0];          // (B, N) float32; d_in[1] = k (static 512)
  float* out_sparse = (float*)d_out;               // first output: (B, N) float32
  int*   out_idx    = (int*)((float*)d_out + (size_t)B_ROWS * N_COLS);  // second: (B, K) int32

  const size_t dyn_lds = (size_t)N_COLS * sizeof(float);  // 64 KB row staging
  topk_sparse_kernel<<<dim3(B_ROWS), dim3(THREADS), dyn_lds, stream>>>(h, out_sparse, out_idx);
}